// Masker_30408368455955
// MI455X (gfx1250) — compile-verified
//
#include <hip/hip_runtime.h>
#include <hip/hip_bf16.h>

typedef __attribute__((ext_vector_type(2))) float v2f;
typedef __attribute__((ext_vector_type(8))) float v8f;

// Problem dims (fixed by reference)
// B=32, C=64, H=W=112, G=4, GRAN=4 -> mh=mw=28
constexpr int NROW   = 32 * 28 * 28;      // 25088 GEMM rows (b, ch, cw)
constexpr int NCOARSE= 32 * 4 * 28 * 28;  // 100352 coarse mask elems
constexpr int NPIX   = 32 * 112 * 112;    // 401408 (b,h,w) pixels
constexpr int NOUT   = 32 * 4 * 112 * 112;// 1605632 elems per big output
constexpr int AST    = 68;                // padded LDS stride (channels)

// ---------------------------------------------------------------------------
// Kernel 1: fused 4x4 average pool + GEMM (M=25088,K=64,N=8 padded to 16)
// via V_WMMA_F32_16X16X4_F32. 196 blocks x 256 threads; each block handles
// 128 rows (8 waves x 16-row tiles). x is read exactly once, coalesced.
// ---------------------------------------------------------------------------
__global__ __launch_bounds__(256) void pool_gemm_kernel(
    const float* __restrict__ x, const float* __restrict__ weight,
    const float* __restrict__ bias, float* __restrict__ logits)
{
    __shared__ float a_lds[128 * AST];  // pooled activations [row][c]
    __shared__ float w_lds[16 * AST];   // weight padded to 16 rows [n][c]

    const int tid   = threadIdx.x;
    const int block = blockIdx.x;

    // Load weight (8x64), zero-pad rows 8..15
    for (int i = tid; i < 16 * 64; i += 256) {
        const int n = i >> 6, c = i & 63;
        w_lds[n * AST + c] = (n < 8) ? weight[n * 64 + c] : 0.0f;
    }

    // Cooperative 4x4 average pooling into LDS.
    // idx = c*128 + row_local  -> consecutive threads hit consecutive cw
    // so each float4 row-segment load is contiguous across the half-wave.
    for (int idx = tid; idx < 64 * 128; idx += 256) {
        const int c  = idx >> 7;
        const int rl = idx & 127;
        const int row = block * 128 + rl;
        const int b  = row / 784;
        const int s  = row % 784;
        const int ch = s / 28, cw = s % 28;
        const float4* p = (const float4*)(x +
            (size_t)(((b * 64 + c) * 112 + ch * 4)) * 112 + cw * 4);
        float sum = 0.0f;
        #pragma unroll
        for (int r = 0; r < 4; ++r) {           // 112 floats = 28 float4
            const float4 v = p[r * 28];
            sum += v.x + v.y + v.z + v.w;
        }
        a_lds[rl * AST + c] = sum * (1.0f / 16.0f);
    }
    __syncthreads();

    // WMMA: one 16x16 tile per wave, K=64 in 16 steps of 4.
    // f32 16x4 A layout: lane m=lane&15, khalf=lane>>4 holds K={k0+2*khalf, +1}
    const int wave  = tid >> 5;
    const int lane  = tid & 31;
    const int m     = lane & 15;          // row-in-tile (A) / col n (B)
    const int khalf = lane >> 4;
    const int rl0   = wave * 16 + m;

    v8f acc = {};
    #pragma unroll
    for (int kk = 0; kk < 16; ++kk) {
        const int k = kk * 4 + khalf * 2;
        const v2f a = *(const v2f*)&a_lds[rl0 * AST + k];
        const v2f bfrag = *(const v2f*)&w_lds[m * AST + k];
        acc = __builtin_amdgcn_wmma_f32_16x16x4_f32(
            false, a, false, bfrag, (short)0, acc, false, false);
    }

    // Epilogue: D vgpr r holds M = r + 8*khalf, col n = lane&15. Only n<8 valid.
    if (m < 8) {
        const float bs = bias[m];
        const int base_row = block * 128 + wave * 16 + khalf * 8;
        #pragma unroll
        for (int r = 0; r < 8; ++r)
            logits[(base_row + r) * 8 + m] = acc[r] + bs;
    }
}

// ---------------------------------------------------------------------------
// Kernel 2: gumbel hard-argmax -> coarse mask (b,g,28,28) + sparsity count
// ---------------------------------------------------------------------------
__global__ __launch_bounds__(256) void mask_kernel(
    const float* __restrict__ logits, const float* __restrict__ gumbel,
    const float* __restrict__ temp, float* __restrict__ maskc,
    unsigned int* __restrict__ acc)
{
    const int i = blockIdx.x * 256 + threadIdx.x;   // grid covers NCOARSE exactly
    const int s = i % 784;
    const int g = (i / 784) & 3;
    const int b = i / (784 * 4);
    const int row = b * 784 + s;
    const float t  = temp[0];
    const float l0 = logits[row * 8 + g];
    const float l1 = logits[row * 8 + 4 + g];
    const float g0 = gumbel[(size_t)((b * 2 + 0) * 4 + g) * 784 + s];
    const float g1 = gumbel[(size_t)((b * 2 + 1) * 4 + g) * 784 + s];
    // argmax over the pair; jnp.argmax picks index 0 on ties -> use >=
    const int mk = ((l0 + g0) / t >= (l1 + g1) / t) ? 1 : 0;
    maskc[i] = (float)mk;
    const unsigned long long bal = __ballot(mk);
    if ((threadIdx.x & 31) == 0)
        atomicAdd(acc, (unsigned int)__popcll(bal));
}

// ---------------------------------------------------------------------------
// Kernel 3: 4x upsample + dilation. dil kernel is all-ones over ALL G input
// channels, so mask_dil is identical across output channels: it's the
// neighborhood-OR (3x3 at fine res -> up to 2x2 coarse cells) of any-g mask.
// ---------------------------------------------------------------------------
__global__ __launch_bounds__(256) void upsample_dilate_kernel(
    const float* __restrict__ maskc, float* __restrict__ out_mask,
    float* __restrict__ out_dil, unsigned int* __restrict__ acc)
{
    const int i = blockIdx.x * 256 + threadIdx.x;   // grid covers NPIX exactly
    const int w = i % 112;
    const int h = (i / 112) % 112;
    const int b = i / (112 * 112);
    const int ch = h >> 2, cw = w >> 2;
    const float* mb = maskc + (size_t)b * 4 * 784;

    #pragma unroll
    for (int g = 0; g < 4; ++g)
        out_mask[(size_t)((b * 4 + g) * 112 + h) * 112 + w] =
            mb[g * 784 + ch * 28 + cw];

    const int chlo = (h == 0)   ? 0  : ((h - 1) >> 2);
    const int chhi = (h == 111) ? 27 : ((h + 1) >> 2);
    const int cwlo = (w == 0)   ? 0  : ((w - 1) >> 2);
    const int cwhi = (w == 111) ? 27 : ((w + 1) >> 2);
    bool any = false;
    for (int g = 0; g < 4 && !any; ++g)
        for (int r = chlo; r <= chhi && !any; ++r)
            for (int cc = cwlo; cc <= cwhi; ++cc)
                if (mb[g * 784 + r * 28 + cc] > 0.5f) { any = true; break; }

    const float dv = any ? 1.0f : 0.0f;
    #pragma unroll
    for (int g = 0; g < 4; ++g)
        out_dil[(size_t)((b * 4 + g) * 112 + h) * 112 + w] = dv;

    const unsigned long long bal = __ballot(any);
    if ((threadIdx.x & 31) == 0)
        atomicAdd(acc + 1, (unsigned int)__popcll(bal));
}

// ---------------------------------------------------------------------------
// Kernel 4: scalars: sparsity, sparsity_dil, flops (constant 501760)
// ---------------------------------------------------------------------------
__global__ void finalize_kernel(const unsigned int* __restrict__ acc,
                                float* __restrict__ scal)
{
    scal[0] = (float)acc[0] / (float)NCOARSE;   // mean of coarse mask
    scal[1] = (float)acc[1] / (float)NPIX;      // identical across 4 channels
    scal[2] = 501760.0f;                        // C*mh*mw + (2G*C + C)*mh*mw
}

extern "C" void kernel_launch(void* const* d_in, const int* in_sizes, int n_in,
                              void* d_out, int out_size, void* d_ws, size_t ws_size,
                              hipStream_t stream)
{
    const float* x      = (const float*)d_in[0];
    const float* weight = (const float*)d_in[1];
    const float* bias   = (const float*)d_in[2];
    const float* gumbel = (const float*)d_in[3];
    const float* temp   = (const float*)d_in[4];
    float* out = (float*)d_out;

    char* ws = (char*)d_ws;
    float* logits = (float*)ws;                                   // 25088*8 f32
    float* maskc  = (float*)(ws + (size_t)NROW * 8 * 4);          // 100352 f32
    unsigned int* acc = (unsigned int*)(ws + (size_t)NROW * 8 * 4
                                           + (size_t)NCOARSE * 4);// 2 u32

    hipMemsetAsync(acc, 0, 2 * sizeof(unsigned int), stream);

    pool_gemm_kernel<<<NROW / 128, 256, 0, stream>>>(x, weight, bias, logits);
    mask_kernel<<<NCOARSE / 256, 256, 0, stream>>>(logits, gumbel, temp, maskc, acc);
    upsample_dilate_kernel<<<NPIX / 256, 256, 0, stream>>>(maskc, out, out + NOUT, acc);
    finalize_kernel<<<1, 1, 0, stream>>>(acc, out + 2 * (size_t)NOUT);
}